// SmoothLoss_45921790329575
// MI455X (gfx1250) — compile-verified
//
#include <hip/hip_runtime.h>
#include <hip/hip_bf16.h>

typedef float v2f __attribute__((ext_vector_type(2)));
typedef float v8f __attribute__((ext_vector_type(8)));

#define HX 512
#define WX 512
#define OHW 513          // gradient-map size (H+1)
#define HY 256           // y spatial size
#define TD 33            // tiles per spatial dim: ceil(513/16)
#define TILES_PER_IMG (TD * TD)
#define NCH 24           // N*C
#define NTILES (NCH * TILES_PER_IMG)

__global__ __launch_bounds__(256) void smooth_loss_main(
    const float* __restrict__ x, const float* __restrict__ y,
    float* __restrict__ partials)
{
    // R halo: rows [ti-2, ti+16], cols [tj-2, tj+16]  -> 19x19 (stride 20)
    __shared__ float Rs[19][20];
    // gradients on [ti-1, ti+16] x [tj-1, tj+16]      -> 18x18 (stride 20)
    __shared__ float Gxs[18][20];
    __shared__ float Gys[18][20];
    __shared__ float wsum[8];

    const int t    = blockIdx.x;
    const int cimg = t / TILES_PER_IMG;
    const int r    = t - cimg * TILES_PER_IMG;
    const int ti   = (r / TD) * 16;
    const int tj   = (r % TD) * 16;
    const int tid  = threadIdx.x;

    const float* __restrict__ xb = x + (size_t)cimg * HX * WX;
    const float* __restrict__ yb = y + (size_t)cimg * HY * HY;

    // ---- Stage 1: bilinear-upsampled R tile (zero outside [0,512)^2) ----
    for (int k = tid; k < 19 * 19; k += 256) {
        int dp = k / 19, dq = k - dp * 19;
        int rp = ti - 2 + dp;
        int cq = tj - 2 + dq;
        float v = 0.0f;
        if (rp >= 0 && rp < HX && cq >= 0 && cq < WX) {
            // src coord = (o + 0.5) * 0.5 - 0.5 = o*0.5 - 0.25, clipped
            float cy = fminf(fmaxf((float)rp * 0.5f - 0.25f, 0.0f), (float)(HY - 1));
            float cx = fminf(fmaxf((float)cq * 0.5f - 0.25f, 0.0f), (float)(HY - 1));
            int y0 = (int)cy; int y1 = min(y0 + 1, HY - 1); float wy = cy - (float)y0;
            int x0 = (int)cx; int x1 = min(x0 + 1, HY - 1); float wx = cx - (float)x0;
            const float* row0 = yb + y0 * HY;
            const float* row1 = yb + y1 * HY;
            float top = row0[x0] + wx * (row0[x1] - row0[x0]);
            float bot = row1[x0] + wx * (row1[x1] - row1[x0]);
            v = top + wy * (bot - top);
        }
        Rs[dp][dq] = v;
    }
    __syncthreads();

    // ---- Stage 2: |grad| maps of R (zero-extension == conv zero padding) ----
    for (int k = tid; k < 18 * 18; k += 256) {
        int dp = k / 18, dq = k - dp * 18;
        float rc = Rs[dp + 1][dq + 1];
        Gxs[dp][dq] = fabsf(rc - Rs[dp + 1][dq]);   // |R(p,q) - R(p,q-1)|
        Gys[dp][dq] = fabsf(rc - Rs[dp][dq + 1]);   // |R(p,q) - R(p-1,q)|
    }
    __syncthreads();

    // ---- Stage 3: per-pixel loss term ----
    const int tx = tid & 15, ty = tid >> 4;
    const int i = ti + ty, j = tj + tx;
    float term = 0.0f;
    if (i < OHW && j < OHW) {
        auto Xv = [&](int ii, int jj) -> float {
            return (ii >= 0 && ii < HX && jj >= 0 && jj < WX)
                       ? xb[ii * WX + jj] : 0.0f;
        };
        float xc = Xv(i, j);
        float Ix = fabsf(xc - Xv(i, j - 1));
        float Iy = fabsf(xc - Xv(i - 1, j));

        float Sx = 0.0f, Sy = 0.0f;
        #pragma unroll
        for (int a = 0; a < 3; ++a) {
            #pragma unroll
            for (int b = 0; b < 3; ++b) {
                Sx += Gxs[ty + a][tx + b];
                Sy += Gys[ty + a][tx + b];
            }
        }
        const float kf = -10.0f / 9.0f;   // exp(-10 * pooled/9)
        term = Ix * __expf(kf * Sx) + Iy * __expf(kf * Sy);
    }

    // ---- Stage 4: wave-level reduction via v_wmma_f32_16x16x4_f32 ----
    // A(16x4): lanes 0-15 -> A[m,0]=term_m (A[m,1]=0); lanes 16-31 -> A[m,2]=term_{16+m}.
    // B = ones(4x16)  =>  D[m,n] = term_m + term_{16+m} for all n.
    v2f a; a[0] = term; a[1] = 0.0f;
    v2f b; b[0] = 1.0f; b[1] = 1.0f;
    v8f c = {0.0f, 0.0f, 0.0f, 0.0f, 0.0f, 0.0f, 0.0f, 0.0f};
    c = __builtin_amdgcn_wmma_f32_16x16x4_f32(
            /*neg_a=*/false, a, /*neg_b=*/false, b,
            /*c_mod=*/(short)0, c, /*reuse_a=*/false, /*reuse_b=*/false);
    float s = c[0] + c[1] + c[2] + c[3] + c[4] + c[5] + c[6] + c[7];
    // lanes<16 hold sum(rows 0..7), lanes>=16 hold sum(rows 8..15); combine halves.
    s += __shfl_xor(s, 16, 32);

    const int lane = tid & 31;
    if (lane == 0) wsum[tid >> 5] = s;
    __syncthreads();
    if (tid == 0) {
        float bs = 0.0f;
        #pragma unroll
        for (int w = 0; w < 8; ++w) bs += wsum[w];
        partials[blockIdx.x] = bs;
    }
}

__global__ __launch_bounds__(256) void smooth_loss_finalize(
    const float* __restrict__ partials, int n, float* __restrict__ out, float scale)
{
    __shared__ float sm[256];
    float s = 0.0f;
    for (int k = threadIdx.x; k < n; k += 256) s += partials[k];
    sm[threadIdx.x] = s;
    __syncthreads();
    for (int off = 128; off > 0; off >>= 1) {
        if (threadIdx.x < off) sm[threadIdx.x] += sm[threadIdx.x + off];
        __syncthreads();
    }
    if (threadIdx.x == 0) out[0] = sm[0] * scale;
}

extern "C" void kernel_launch(void* const* d_in, const int* in_sizes, int n_in,
                              void* d_out, int out_size, void* d_ws, size_t ws_size,
                              hipStream_t stream) {
    const float* x = (const float*)d_in[0];   // (8,3,512,512) f32
    const float* y = (const float*)d_in[1];   // (8,3,256,256) f32
    float* out      = (float*)d_out;          // scalar f32
    float* partials = (float*)d_ws;           // NTILES floats (fully overwritten)

    smooth_loss_main<<<NTILES, 256, 0, stream>>>(x, y, partials);
    // mean over 24*513*513 = 6,316,056 elements
    smooth_loss_finalize<<<1, 256, 0, stream>>>(partials, NTILES, out,
                                                1.0f / 6316056.0f);
}